// MomentsToImage_39075612459061
// MI455X (gfx1250) — compile-verified
//
#include <hip/hip_runtime.h>

typedef float v2f __attribute__((ext_vector_type(2)));
typedef float v4f __attribute__((ext_vector_type(4)));
typedef float v8f __attribute__((ext_vector_type(8)));

#define IMG_H 128
#define IMG_W 128

// One block per batch element; 8 waves per block; wave w owns the 16-row
// stripe y in [16w, 16w+16) and loops over the 8 x-tiles.
// Per 16x16 tile: 1x v_wmma_f32_16x16x4_f32 computes the full log2-space
// Gaussian exponent (rank-3 bilinear form + constant folded into K=0..2),
// then 8x v_exp_f32 and two NT b128 stores per lane.
__global__ __launch_bounds__(256) void moments_to_image_kernel(
    const float* __restrict__ moments, float* __restrict__ out) {
  const int b    = blockIdx.x;
  const int wave = threadIdx.x >> 5;   // 0..7  -> tile row
  const int lane = threadIdx.x & 31;
  const int hf   = lane >> 4;          // which lane-half (K-half / M-half)
  const int l    = lane & 15;

  // ---- per-batch coefficients (block-uniform -> scalar loads, done once) ----
  const float sx = (float)(IMG_W - 1);     // 127
  const float sy = (float)(IMG_H - 1);     // 127
  const float m0 = moments[b * 6 + 0];
  const float m1 = moments[b * 6 + 1];
  const float m2 = moments[b * 6 + 2];
  const float m3 = moments[b * 6 + 3];
  const float m4 = moments[b * 6 + 4];
  const float m5 = moments[b * 6 + 5];

  const float mux = tanhf(m0) * (sx * 0.5f) + (sx * 0.5f);
  const float muy = tanhf(m1) * (sy * 0.5f) + (sy * 0.5f);

  const float t0 = fminf(fmaxf(tanhf(m2),  0.03f ), 1.0f );
  const float t1 = fminf(fmaxf(tanhf(m3), -0.999f), 0.999f);
  const float t2 = fminf(fmaxf(tanhf(m4),  0.03f ), 1.0f );

  const float ca  = t0 * 3.0f * sx;                               // cov[0,0]
  const float cb  = t1 * sqrtf(t0 * t2 * 9.0f * sx * sy);         // cov[0,1]
  const float cc  = t2 * 3.0f * sy;                               // cov[1,1]
  const float det = ca * cc - cb * cb;                            // > 0

  const float it       = 10.0f / (1.0f + __expf(-m5));            // 10*sigmoid
  const float norm     = it * rsqrtf(det * 39.478417604357434f);  // /(2pi)^2
  const float log2norm = __log2f(norm);

  const float s2  = -0.72134752044448170f / det;  // -0.5*log2(e)/det
  const float kxx = s2 * cc;                      //  * dx^2
  const float kxy = -2.0f * cb * s2;              //  * dx*dy
  const float kyy = s2 * ca;                      //  * dy^2

  // ---- B operand: depends only on y -> fixed for the whole wave ----
  // B(4x16) f32 layout: V0 = {lanes0-15: K=0 row, lanes16-31: K=2 row},
  //                     V1 = {lanes0-15: K=1 row, lanes16-31: K=3 row}, N = lane%16
  const int   y0 = wave * 16;
  const float dy = (float)(y0 + l) - muy;
  v2f Bv;
  Bv.x = hf ? (kyy * dy * dy) : 1.0f;   // K=2 : s*a*dy^2   | K=0 : 1
  Bv.y = hf ? 0.0f            : dy;     // K=3 : 0          | K=1 : dy

  // D(16x16) f32 layout: lane<16 v[k] = D[M=k][N=lane]; lane>=16 v[k] = D[M=8+k][N=lane-16]
  // -> each lane stores 8 consecutive x at fixed y = y0 + l, x base = x0 + hf*8.
  float* rowbase = out + (size_t)b * (IMG_H * IMG_W) + (size_t)(y0 + l) * IMG_W + hf * 8;

  for (int tx = 0; tx < 8; ++tx) {
    const int   x0 = tx * 16;
    const float dx = (float)(x0 + l) - mux;

    // A operand: A(16x4) f32 layout: lanes0-15 V0/V1 = K=0/K=1 of row M=lane,
    //            lanes16-31 V0/V1 = K=2/K=3 of row M=lane-16.
    v2f Av;
    Av.x = hf ? 1.0f : (kxx * dx * dx + log2norm);  // K=0 : s*c*dx^2+log2(norm) | K=2 : 1
    Av.y = hf ? 0.0f : (kxy * dx);                  // K=1 : -2b*s*dx            | K=3 : 0

    v8f acc = {};
    acc = __builtin_amdgcn_wmma_f32_16x16x4_f32(
        /*neg_a=*/false, Av, /*neg_b=*/false, Bv,
        /*c_mod=*/(short)0, acc, /*reuse_a=*/false, /*reuse_b=*/false);

    v4f r0, r1;
    r0.x = __builtin_amdgcn_exp2f(acc[0]);
    r0.y = __builtin_amdgcn_exp2f(acc[1]);
    r0.z = __builtin_amdgcn_exp2f(acc[2]);
    r0.w = __builtin_amdgcn_exp2f(acc[3]);
    r1.x = __builtin_amdgcn_exp2f(acc[4]);
    r1.y = __builtin_amdgcn_exp2f(acc[5]);
    r1.z = __builtin_amdgcn_exp2f(acc[6]);
    r1.w = __builtin_amdgcn_exp2f(acc[7]);

    // 256 MB streamed once, never re-read: non-temporal b128 stores.
    v4f* p = (v4f*)(rowbase + x0);
    __builtin_nontemporal_store(r0, p);
    __builtin_nontemporal_store(r1, p + 1);
  }
}

extern "C" void kernel_launch(void* const* d_in, const int* in_sizes, int n_in,
                              void* d_out, int out_size, void* d_ws, size_t ws_size,
                              hipStream_t stream) {
  const float* moments = (const float*)d_in[0];
  float* out = (float*)d_out;
  const int B = in_sizes[0] / 6;  // 4096
  moments_to_image_kernel<<<dim3(B), dim3(256), 0, stream>>>(moments, out);
}